// DGraphDTA_36799279792821
// MI455X (gfx1250) — compile-verified
//
#include <hip/hip_runtime.h>
#include <hip/hip_bf16.h>

// ---------------------------------------------------------------------------
// Types for CDNA5 WMMA f32 16x16x4:
//   A (16x4 f32)  = 2 VGPRs -> v2f ; B (4x16 f32) = 2 VGPRs -> v2f
//   C/D (16x16)   = 8 VGPRs -> v8f
// ---------------------------------------------------------------------------
typedef __attribute__((ext_vector_type(2))) float v2f;
typedef __attribute__((ext_vector_type(8))) float v8f;

// Problem constants (from reference setup_inputs)
#define NB        256
#define N_MOL     10240
#define E_MOL     40960
#define F_MOL     78
#define N_PRO     76800
#define E_PRO     768000
#define F_PRO     54
#define OUT_DIM   128

// ---------------------------------------------------------------------------
// fp32 WMMA GEMM: C[M,N] = A[M,K] @ B[K,N] (+bias, optional relu)
//
// Tiling: 4 waves / block (blockDim = 32x4). Each wave computes a 64x16 C
// slab (4 M-tiles stacked) so the B (weight) fragment is loaded once per
// K-step and fed to 4 WMMAs. A block covers 256 rows x 16 cols.
//
// REQUIREMENT: K must be even (true for every GEMM in this model). This lets
// the (ka, ka+1) fragment pair be all-valid or all-invalid, so out-of-range
// handling is done with clamped addresses + value selects (v_cndmask) —
// no divergent branches, EXEC stays all-ones through the WMMAs.
//
// Fragment layouts per CDNA5 ISA 7.12.2:
//   A: lane L (half=L>>4, m=L&15): v0 = A[m][ka], v1 = A[m][ka+1], ka=k0+2*half
//   B: lane L (half, n=L&15):      v0 = B[ka][n], v1 = B[ka+1][n]
//   C: VGPR i: lanes 0-15 -> (M=i, N=lane); lanes 16-31 -> (M=8+i, N=lane-16)
// ---------------------------------------------------------------------------
#define WAVE_ROWS  64
#define BLOCK_ROWS 256

__global__ __launch_bounds__(128) void wmma_gemm_f32(
    const float* __restrict__ A, int lda,
    const float* __restrict__ B, int ldb,
    const float* __restrict__ bias,
    float* __restrict__ C, int ldc,
    int M, int N, int K, int relu)
{
    const int tileM0 = blockIdx.x * BLOCK_ROWS + threadIdx.y * WAVE_ROWS;
    const int tileN  = blockIdx.y * 16;
    const int lane   = threadIdx.x;       // 0..31, wave32
    const int half   = lane >> 4;         // 0 or 1
    const int l16    = lane & 15;

    const int n  = tileN + l16;
    const int nc = (n < N) ? n : (N - 1);       // clamped column
    const bool nok = (n < N);

    // Clamped A rows for the 4 stacked M-tiles
    int mrow[4]; bool mok[4];
    const float* arow[4];
#pragma unroll
    for (int t = 0; t < 4; ++t) {
        int m  = tileM0 + t * 16 + l16;
        mok[t]  = (m < M);
        mrow[t] = mok[t] ? m : (M - 1);
        arow[t] = A + (size_t)mrow[t] * lda;
    }

    v8f acc[4] = {};
    for (int k0 = 0; k0 < K; k0 += 4) {
        const int ka  = k0 + 2 * half;          // even
        const bool kok = (ka < K);              // K even => ka+1 < K too
        const int kc  = kok ? ka : (K - 2);     // clamped, even

        // B fragment (shared across the 4 WMMAs)
        float b0 = B[(size_t)kc * ldb + nc];
        float b1 = B[(size_t)(kc + 1) * ldb + nc];
        v2f bf;
        const bool bok = nok && kok;
        bf.x = bok ? b0 : 0.0f;
        bf.y = bok ? b1 : 0.0f;

#pragma unroll
        for (int t = 0; t < 4; ++t) {
            // A fragment: contiguous pair -> single 8-byte load
            v2f av = *(const v2f*)(arow[t] + kc);
            v2f af;
            const bool aok = kok && mok[t];
            af.x = aok ? av.x : 0.0f;
            af.y = aok ? av.y : 0.0f;
            acc[t] = __builtin_amdgcn_wmma_f32_16x16x4_f32(
                false, af, false, bf, (short)0, acc[t], false, false);
        }
    }

    if (nok) {
        const float bv = bias ? bias[n] : 0.0f;
#pragma unroll
        for (int t = 0; t < 4; ++t) {
#pragma unroll
            for (int i = 0; i < 8; ++i) {
                const int mr = tileM0 + t * 16 + 8 * half + i;
                if (mr < M) {
                    float v = acc[t][i] + bv;
                    if (relu) v = fmaxf(v, 0.0f);
                    C[(size_t)mr * ldc + n] = v;
                }
            }
        }
    }
}

static inline dim3 gemm_grid(int M, int N) {
    return dim3((unsigned)((M + BLOCK_ROWS - 1) / BLOCK_ROWS),
                (unsigned)((N + 15) / 16), 1);
}
static inline dim3 gemm_block() { return dim3(32, 4, 1); }

// ---------------------------------------------------------------------------
// Elementwise / scatter kernels
// ---------------------------------------------------------------------------
__global__ void zero_f32(float* __restrict__ p, int n) {
    int i = blockIdx.x * blockDim.x + threadIdx.x;
    if (i < n) p[i] = 0.0f;
}

__global__ void deg_accum(const int* __restrict__ dst, float* __restrict__ deg, int E) {
    int e = blockIdx.x * blockDim.x + threadIdx.x;
    if (e < E) atomicAdd(&deg[dst[e]], 1.0f);
}

__global__ void finalize_norm(float* __restrict__ deg, int n) {
    int i = blockIdx.x * blockDim.x + threadIdx.x;
    if (i < n) deg[i] = rsqrtf(deg[i] + 1.0f);   // +1 self-loop
}

// agg[dst] += h[src] * norm[src]   (one block per edge, threads over features)
__global__ void scatter_edges(const float* __restrict__ h, const float* __restrict__ norm,
                              const int* __restrict__ src, const int* __restrict__ dst,
                              float* __restrict__ agg, int E, int F)
{
    int e = blockIdx.x;
    if (e >= E) return;
    int s = src[e], d = dst[e];
    float ns = norm[s];
    const float* hs = h + (size_t)s * F;
    float* ad = agg + (size_t)d * F;
    for (int f = threadIdx.x; f < F; f += blockDim.x)
        atomicAdd(&ad[f], hs[f] * ns);
}

// out = relu( norm*(agg + norm*h) + bias )
__global__ void gcn_combine(const float* __restrict__ h, const float* __restrict__ agg,
                            const float* __restrict__ norm, const float* __restrict__ bias,
                            float* __restrict__ out, int Nn, int F)
{
    int i = blockIdx.x * blockDim.x + threadIdx.x;
    int total = Nn * F;
    if (i >= total) return;
    int node = i / F;
    int f = i - node * F;
    float nv = norm[node];
    float v = nv * (agg[i] + nv * h[i]) + bias[f];
    out[i] = fmaxf(v, 0.0f);
}

// segment-sum pooling: sums[batch[node]] += x[node], counts[batch[node]] += 1
__global__ void pool_accum(const float* __restrict__ x, const int* __restrict__ batch,
                           float* __restrict__ sums, float* __restrict__ counts,
                           int Nn, int F)
{
    int node = blockIdx.x;
    if (node >= Nn) return;
    int b = batch[node];
    if (threadIdx.x == 0) atomicAdd(&counts[b], 1.0f);
    const float* xn = x + (size_t)node * F;
    float* sb = sums + (size_t)b * F;
    for (int f = threadIdx.x; f < F; f += blockDim.x)
        atomicAdd(&sb[f], xn[f]);
}

__global__ void pool_div(float* __restrict__ sums, const float* __restrict__ counts,
                         int Bn, int F)
{
    int i = blockIdx.x * blockDim.x + threadIdx.x;
    int total = Bn * F;
    if (i >= total) return;
    int b = i / F;
    sums[i] /= fmaxf(counts[b], 1.0f);
}

// ---------------------------------------------------------------------------
// Host-side helpers (all launches on `stream`, no sync / alloc)
// ---------------------------------------------------------------------------
static inline void launch_zero(float* p, int n, hipStream_t s) {
    zero_f32<<<(n + 255) / 256, 256, 0, s>>>(p, n);
}

static void gcn_layer(const float* x, int Nn, int Fin, int Fout,
                      const float* W, const float* b,
                      const int* src, const int* dst, int E,
                      const float* norm,
                      float* hbuf, float* aggbuf, float* out, hipStream_t stream)
{
    // h = x @ W
    wmma_gemm_f32<<<gemm_grid(Nn, Fout), gemm_block(), 0, stream>>>(
        x, Fin, W, Fout, nullptr, hbuf, Fout, Nn, Fout, Fin, 0);
    // agg = segment_sum(h[src]*norm[src], dst)
    launch_zero(aggbuf, Nn * Fout, stream);
    scatter_edges<<<E, 64, 0, stream>>>(hbuf, norm, src, dst, aggbuf, E, Fout);
    // out = relu(norm*(agg + norm*h) + b)
    gcn_combine<<<(Nn * Fout + 255) / 256, 256, 0, stream>>>(
        hbuf, aggbuf, norm, b, out, Nn, Fout);
}

extern "C" void kernel_launch(void* const* d_in, const int* in_sizes, int n_in,
                              void* d_out, int out_size, void* d_ws, size_t ws_size,
                              hipStream_t stream)
{
    (void)in_sizes; (void)n_in; (void)out_size; (void)ws_size;

    // ---- inputs -----------------------------------------------------------
    const float* mol_x   = (const float*)d_in[0];
    const int*   mol_ei  = (const int*)d_in[1];
    const int*   mol_bat = (const int*)d_in[2];
    const float* pro_x   = (const float*)d_in[3];
    const int*   pro_ei  = (const int*)d_in[4];
    const int*   pro_bat = (const int*)d_in[5];

    const float* mw1 = (const float*)d_in[6];  const float* mb1 = (const float*)d_in[7];
    const float* mw2 = (const float*)d_in[8];  const float* mb2 = (const float*)d_in[9];
    const float* mw3 = (const float*)d_in[10]; const float* mb3 = (const float*)d_in[11];
    const float* mfw1 = (const float*)d_in[12]; const float* mfb1 = (const float*)d_in[13];
    const float* mfw2 = (const float*)d_in[14]; const float* mfb2 = (const float*)d_in[15];

    const float* pw1 = (const float*)d_in[16]; const float* pb1 = (const float*)d_in[17];
    const float* pw2 = (const float*)d_in[18]; const float* pb2 = (const float*)d_in[19];
    const float* pw3 = (const float*)d_in[20]; const float* pb3 = (const float*)d_in[21];
    const float* pfw1 = (const float*)d_in[22]; const float* pfb1 = (const float*)d_in[23];
    const float* pfw2 = (const float*)d_in[24]; const float* pfb2 = (const float*)d_in[25];

    const float* fc1w = (const float*)d_in[26]; const float* fc1b = (const float*)d_in[27];
    const float* fc2w = (const float*)d_in[28]; const float* fc2b = (const float*)d_in[29];
    const float* outw = (const float*)d_in[30]; const float* outb = (const float*)d_in[31];

    const int* mol_src = mol_ei;            const int* mol_dst = mol_ei + E_MOL;
    const int* pro_src = pro_ei;            const int* pro_dst = pro_ei + E_PRO;

    float* out = (float*)d_out;             // [256, 1]

    // ---- workspace layout (floats) ---------------------------------------
    const size_t BIG = (size_t)N_PRO * (4 * F_PRO);   // 76800*216, >= 10240*312
    float* ws = (float*)d_ws;
    float* feat  = ws;                  // current node features
    float* hbuf  = feat  + BIG;         // h = x @ W
    float* aggbf = hbuf  + BIG;         // scattered aggregate
    float* normb = aggbf + BIG;         // per-node norm (max N_PRO)
    float* poolM = normb + N_PRO;                  // [256, 312]
    float* poolP = poolM + (size_t)NB * 4 * F_MOL; // [256, 216]
    float* cnts  = poolP + (size_t)NB * 4 * F_PRO; // [256]
    float* fcbuf = cnts  + NB;                     // [256, 1024]
    float* ccat  = fcbuf + (size_t)NB * 1024;      // [256, 256] concat
    float* fc2o  = ccat  + (size_t)NB * 2 * OUT_DIM; // [256, 512]

    // ======================= molecule branch ==============================
    launch_zero(normb, N_MOL, stream);
    deg_accum<<<(E_MOL + 255) / 256, 256, 0, stream>>>(mol_dst, normb, E_MOL);
    finalize_norm<<<(N_MOL + 255) / 256, 256, 0, stream>>>(normb, N_MOL);

    gcn_layer(mol_x, N_MOL, F_MOL,     F_MOL,     mw1, mb1, mol_src, mol_dst, E_MOL, normb, hbuf, aggbf, feat, stream);
    gcn_layer(feat,  N_MOL, F_MOL,     2 * F_MOL, mw2, mb2, mol_src, mol_dst, E_MOL, normb, hbuf, aggbf, feat, stream);
    gcn_layer(feat,  N_MOL, 2 * F_MOL, 4 * F_MOL, mw3, mb3, mol_src, mol_dst, E_MOL, normb, hbuf, aggbf, feat, stream);

    launch_zero(poolM, NB * 4 * F_MOL, stream);
    launch_zero(cnts, NB, stream);
    pool_accum<<<N_MOL, 64, 0, stream>>>(feat, mol_bat, poolM, cnts, N_MOL, 4 * F_MOL);
    pool_div<<<(NB * 4 * F_MOL + 255) / 256, 256, 0, stream>>>(poolM, cnts, NB, 4 * F_MOL);

    // FC head of molecule branch: relu(pool @ mfw1 + mfb1) @ mfw2 + mfb2
    wmma_gemm_f32<<<gemm_grid(NB, 1024), gemm_block(), 0, stream>>>(
        poolM, 4 * F_MOL, mfw1, 1024, mfb1, fcbuf, 1024, NB, 1024, 4 * F_MOL, 1);
    // write into concat columns [0,128)
    wmma_gemm_f32<<<gemm_grid(NB, OUT_DIM), gemm_block(), 0, stream>>>(
        fcbuf, 1024, mfw2, OUT_DIM, mfb2, ccat, 2 * OUT_DIM, NB, OUT_DIM, 1024, 0);

    // ======================= protein branch ===============================
    launch_zero(normb, N_PRO, stream);
    deg_accum<<<(E_PRO + 255) / 256, 256, 0, stream>>>(pro_dst, normb, E_PRO);
    finalize_norm<<<(N_PRO + 255) / 256, 256, 0, stream>>>(normb, N_PRO);

    gcn_layer(pro_x, N_PRO, F_PRO,     F_PRO,     pw1, pb1, pro_src, pro_dst, E_PRO, normb, hbuf, aggbf, feat, stream);
    gcn_layer(feat,  N_PRO, F_PRO,     2 * F_PRO, pw2, pb2, pro_src, pro_dst, E_PRO, normb, hbuf, aggbf, feat, stream);
    gcn_layer(feat,  N_PRO, 2 * F_PRO, 4 * F_PRO, pw3, pb3, pro_src, pro_dst, E_PRO, normb, hbuf, aggbf, feat, stream);

    launch_zero(poolP, NB * 4 * F_PRO, stream);
    launch_zero(cnts, NB, stream);
    pool_accum<<<N_PRO, 64, 0, stream>>>(feat, pro_bat, poolP, cnts, N_PRO, 4 * F_PRO);
    pool_div<<<(NB * 4 * F_PRO + 255) / 256, 256, 0, stream>>>(poolP, cnts, NB, 4 * F_PRO);

    wmma_gemm_f32<<<gemm_grid(NB, 1024), gemm_block(), 0, stream>>>(
        poolP, 4 * F_PRO, pfw1, 1024, pfb1, fcbuf, 1024, NB, 1024, 4 * F_PRO, 1);
    // write into concat columns [128,256)
    wmma_gemm_f32<<<gemm_grid(NB, OUT_DIM), gemm_block(), 0, stream>>>(
        fcbuf, 1024, pfw2, OUT_DIM, pfb2, ccat + OUT_DIM, 2 * OUT_DIM, NB, OUT_DIM, 1024, 0);

    // ======================= combined head ================================
    // relu(ccat @ fc1w + fc1b) -> fcbuf [256,1024]
    wmma_gemm_f32<<<gemm_grid(NB, 1024), gemm_block(), 0, stream>>>(
        ccat, 2 * OUT_DIM, fc1w, 1024, fc1b, fcbuf, 1024, NB, 1024, 2 * OUT_DIM, 1);
    // relu(fcbuf @ fc2w + fc2b) -> fc2o [256,512]
    wmma_gemm_f32<<<gemm_grid(NB, 512), gemm_block(), 0, stream>>>(
        fcbuf, 1024, fc2w, 512, fc2b, fc2o, 512, NB, 512, 1024, 1);
    // fc2o @ outw + outb -> out [256,1]
    wmma_gemm_f32<<<gemm_grid(NB, 1), gemm_block(), 0, stream>>>(
        fc2o, 512, outw, 1, outb, out, 1, NB, 1, 512, 0);
}